// Encoder_3848290697639
// MI455X (gfx1250) — compile-verified
//
#include <hip/hip_runtime.h>

// ---------------------------------------------------------------------------
// GAT encoder for MI455X (gfx1250, wave32).
// Edge phase: 3-pass stable segment-softmax with L2-resident f32 atomics
//             (all gather tables < 80MB << 192MB L2).
// Dense phase: V_WMMA_F32_16X16X4_F32 tiles (full f32 precision; GEMMs are
//              not the bottleneck so no reason to drop to f16).
// ---------------------------------------------------------------------------

#define D1C 64    // GAT layer-1 hidden
#define HC  128   // hidden dim

typedef float v2f __attribute__((ext_vector_type(2)));
typedef float v8f __attribute__((ext_vector_type(8)));

__device__ __forceinline__ float lrelu02(float v) { return v > 0.0f ? v : 0.2f * v; }

// float atomic max via signed/unsigned int ordering trick (init must be -inf)
__device__ __forceinline__ void atomicMaxF(float* addr, float val) {
  if (val >= 0.0f) atomicMax((int*)addr, __float_as_int(val));
  else             atomicMin((unsigned int*)addr, (unsigned int)__float_as_int(val));
}

__global__ __launch_bounds__(256) void fill_f32(float* __restrict__ p, int n, float v) {
  int i = blockIdx.x * 256 + threadIdx.x;
  if (i < n) p[i] = v;
}

// Layer-1 feature transform is rank-1: h1[i,k] = x[i] * W1[k]
__global__ __launch_bounds__(256) void l1_rank1_feat(const float* __restrict__ x,
    const float* __restrict__ W1, float* __restrict__ h1, int n) {
  int gid = blockIdx.x * 256 + threadIdx.x;
  if (gid >= n * D1C) return;
  h1[gid] = x[gid >> 6] * W1[gid & (D1C - 1)];
}

// Per-node attention logits: aS[i] = h[i,:]·a_src, aD[i] = h[i,:]·a_dst
// One wave32 per node; coalesced row load + shuffle reduction.
__global__ __launch_bounds__(256) void node_alpha(const float* __restrict__ h,
    const float* __restrict__ asrc, const float* __restrict__ adst,
    float* __restrict__ aS, float* __restrict__ aD, int n, int D) {
  int node = (blockIdx.x * 256 + threadIdx.x) >> 5;
  int lane = threadIdx.x & 31;
  if (node >= n) return;
  const float* row = h + (size_t)node * D;
  float s = 0.0f, d = 0.0f;
  for (int k = lane; k < D; k += 32) {
    float v = row[k];
    s += v * asrc[k];
    d += v * adst[k];
  }
#pragma unroll
  for (int off = 16; off; off >>= 1) {
    s += __shfl_xor(s, off, 32);
    d += __shfl_xor(d, off, 32);
  }
  if (lane == 0) { aS[node] = s; aD[node] = d; }
}

// Pass 1: per-dst running max of e = leaky_relu(aS[s]+aD[d]).  i>=ne => self loop.
__global__ __launch_bounds__(256) void edge_softmax_max(const int* __restrict__ src,
    const int* __restrict__ dst, const float* __restrict__ aS, const float* __restrict__ aD,
    float* __restrict__ mx, int ne, int n) {
  int i = blockIdx.x * 256 + threadIdx.x;
  if (i >= ne + n) return;
  int s, d;
  if (i < ne) { s = src[i]; d = dst[i]; } else { s = d = i - ne; }
  atomicMaxF(&mx[d], lrelu02(aS[s] + aD[d]));
}

// Pass 2: z[d] += exp(e - m[d])
__global__ __launch_bounds__(256) void edge_softmax_sum(const int* __restrict__ src,
    const int* __restrict__ dst, const float* __restrict__ aS, const float* __restrict__ aD,
    const float* __restrict__ mx, float* __restrict__ zz, int ne, int n) {
  int i = blockIdx.x * 256 + threadIdx.x;
  if (i >= ne + n) return;
  int s, d;
  if (i < ne) { s = src[i]; d = dst[i]; } else { s = d = i - ne; }
  float e = lrelu02(aS[s] + aD[d]);
  atomicAdd(&zz[d], __expf(e - mx[d]));
}

// Pass 3: agg[d,:] += attn * h[s,:].  D consecutive threads handle one edge,
// so gathers/atomics are coalesced; attn recomputed per-thread (cheap VALU).
__global__ __launch_bounds__(256) void edge_aggregate(const int* __restrict__ src,
    const int* __restrict__ dst, const float* __restrict__ aS, const float* __restrict__ aD,
    const float* __restrict__ mx, const float* __restrict__ zz,
    const float* __restrict__ h, float* __restrict__ agg, int ne, int n, int logD) {
  long gid = (long)blockIdx.x * 256 + threadIdx.x;
  const int D = 1 << logD;
  const long total = (long)(ne + n) << logD;
  if (gid >= total) return;
  int e = (int)(gid >> logD);
  int k = (int)gid & (D - 1);
  int s, d;
  if (e < ne) { s = src[e]; d = dst[e]; } else { s = d = e - ne; }
  float ev = lrelu02(aS[s] + aD[d]);
  float attn = __expf(ev - mx[d]) / zz[d];            // zz > 0 guaranteed by self loop
  atomicAdd(&agg[(size_t)d * D + k], attn * h[(size_t)s * D + k]);
}

// h2[N,128] = relu(agg1 + b1) @ W2^T  via V_WMMA_F32_16X16X4_F32.
// Block = 16-row tile x 128 cols; 8 waves, one 16x16 tile each; K=64 -> 16 WMMA.
__global__ __launch_bounds__(256) void gat2_transform_wmma(const float* __restrict__ agg1,
    const float* __restrict__ b1, const float* __restrict__ W2,
    float* __restrict__ h2, int n) {
  constexpr int K = D1C;
  __shared__ float As[16][K + 1];                     // +1 pad: conflict-free column reads
  const int rowbase = blockIdx.x * 16;
  const int tid = threadIdx.x;
  for (int i = tid; i < 16 * K; i += 256) {
    int r = i >> 6, k = i & (K - 1);
    int gr = rowbase + r;
    float v = (gr < n) ? (agg1[(size_t)gr * K + k] + b1[k]) : 0.0f;
    As[r][k] = v > 0.0f ? v : 0.0f;                   // fused bias + ReLU
  }
  __syncthreads();
  const int lane = tid & 31;
  const int n0 = (tid >> 5) * 16;                     // wave's 16-col tile
  const int m = lane & 15;
  const int kh = (lane >> 4) * 2;                     // 0 | 2 : K-slot per lane half
  const float* wrow = W2 + (size_t)(n0 + m) * K;      // B[k][n] = W2[n0+n][k]
  v8f acc = {};
#pragma unroll
  for (int k0 = 0; k0 < K; k0 += 4) {
    v2f a, b;
    a.x = As[m][k0 + kh]; a.y = As[m][k0 + kh + 1];
    b.x = wrow[k0 + kh];  b.y = wrow[k0 + kh + 1];
    acc = __builtin_amdgcn_wmma_f32_16x16x4_f32(false, a, false, b, (short)0, acc,
                                                false, false);
  }
  const int rowoff = (lane >> 4) * 8;                 // D layout: row = v + 8*hi
#pragma unroll
  for (int v = 0; v < 8; ++v) {
    int gr = rowbase + rowoff + v;
    if (gr < n) h2[(size_t)gr * HC + n0 + m] = acc[v];
  }
}

// out[i,:] = x[i]*Wl1 + bl1 + relu((agg2 + b_gat2) @ Wl2^T + bl2)
// agg2 lives in d_out and is overwritten in place: each block reads only its own
// 16-row tile into LDS before the barrier, then stores after — no cross-block overlap.
__global__ __launch_bounds__(256) void final_fused_wmma(const float* __restrict__ x,
    const float* __restrict__ bg2, const float* __restrict__ Wl1,
    const float* __restrict__ bl1, const float* __restrict__ Wl2,
    const float* __restrict__ bl2, float* __restrict__ out, int n) {
  constexpr int K = HC;
  __shared__ float As[16][K + 1];
  const int rowbase = blockIdx.x * 16;
  const int tid = threadIdx.x;
  for (int i = tid; i < 16 * K; i += 256) {
    int r = i >> 7, k = i & (K - 1);
    int gr = rowbase + r;
    As[r][k] = (gr < n) ? (out[(size_t)gr * K + k] + bg2[k]) : 0.0f;  // fused GAT2 bias
  }
  __syncthreads();
  const int lane = tid & 31;
  const int n0 = (tid >> 5) * 16;
  const int m = lane & 15;
  const int kh = (lane >> 4) * 2;
  const float* wrow = Wl2 + (size_t)(n0 + m) * K;
  v8f acc = {};
#pragma unroll
  for (int k0 = 0; k0 < K; k0 += 4) {
    v2f a, b;
    a.x = As[m][k0 + kh]; a.y = As[m][k0 + kh + 1];
    b.x = wrow[k0 + kh];  b.y = wrow[k0 + kh + 1];
    acc = __builtin_amdgcn_wmma_f32_16x16x4_f32(false, a, false, b, (short)0, acc,
                                                false, false);
  }
  const int c = n0 + m;
  const float w1 = Wl1[c], bb1 = bl1[c], bb2 = bl2[c];
  const int rowoff = (lane >> 4) * 8;
#pragma unroll
  for (int v = 0; v < 8; ++v) {
    int gr = rowbase + rowoff + v;
    if (gr < n) {
      float t = acc[v] + bb2;                          // x1 + b_lin2
      out[(size_t)gr * HC + c] = fmaf(x[gr], w1, bb1) + (t > 0.0f ? t : 0.0f);
    }
  }
}

extern "C" void kernel_launch(void* const* d_in, const int* in_sizes, int n_in,
                              void* d_out, int out_size, void* d_ws, size_t ws_size,
                              hipStream_t stream) {
  const float* x   = (const float*)d_in[0];
  const int*   ei  = (const int*)d_in[1];
  const float* W1  = (const float*)d_in[2];
  const float* as1 = (const float*)d_in[3];
  const float* ad1 = (const float*)d_in[4];
  const float* b1  = (const float*)d_in[5];
  const float* W2  = (const float*)d_in[6];
  const float* as2 = (const float*)d_in[7];
  const float* ad2 = (const float*)d_in[8];
  const float* b2  = (const float*)d_in[9];
  const float* Wl1 = (const float*)d_in[10];
  const float* bl1 = (const float*)d_in[11];
  const float* Wl2 = (const float*)d_in[12];
  const float* bl2 = (const float*)d_in[13];
  float* out = (float*)d_out;

  const int n  = in_sizes[0];        // 100000 nodes
  const int ne = in_sizes[1] / 2;    // 600000 edges
  const int* srcp = ei;
  const int* dstp = ei + ne;

  // Workspace layout (floats): 104 MB total; agg2 aliases d_out.
  float* ws   = (float*)d_ws;
  float* h1   = ws;                          // [n,64]
  float* agg1 = h1 + (size_t)n * D1C;        // [n,64]
  float* h2   = agg1 + (size_t)n * D1C;      // [n,128]
  float* aS   = h2 + (size_t)n * HC;         // [n]
  float* aD   = aS + n;                      // [n]
  float* mx   = aD + n;                      // [n]
  float* zz   = mx + n;                      // [n]

  auto cdiv = [](long a, long b) { return (int)((a + b - 1) / b); };
  const float NEG_INF = -__builtin_inff();
  const int em_blocks = cdiv(ne + n, 256);

  // ---- GAT layer 1 (1 -> 64) ----
  fill_f32<<<cdiv(n, 256), 256, 0, stream>>>(mx, n, NEG_INF);
  fill_f32<<<cdiv(n, 256), 256, 0, stream>>>(zz, n, 0.0f);
  fill_f32<<<cdiv((long)n * D1C, 256), 256, 0, stream>>>(agg1, n * D1C, 0.0f);
  l1_rank1_feat<<<cdiv((long)n * D1C, 256), 256, 0, stream>>>(x, W1, h1, n);
  node_alpha<<<cdiv((long)n * 32, 256), 256, 0, stream>>>(h1, as1, ad1, aS, aD, n, D1C);
  edge_softmax_max<<<em_blocks, 256, 0, stream>>>(srcp, dstp, aS, aD, mx, ne, n);
  edge_softmax_sum<<<em_blocks, 256, 0, stream>>>(srcp, dstp, aS, aD, mx, zz, ne, n);
  edge_aggregate<<<cdiv((long)(ne + n) << 6, 256), 256, 0, stream>>>(
      srcp, dstp, aS, aD, mx, zz, h1, agg1, ne, n, 6);

  // ---- GAT layer 2 feature transform (relu + bias fused, WMMA) ----
  gat2_transform_wmma<<<cdiv(n, 16), 256, 0, stream>>>(agg1, b1, W2, h2, n);

  // ---- GAT layer 2 attention (agg2 accumulated directly in d_out) ----
  fill_f32<<<cdiv(n, 256), 256, 0, stream>>>(mx, n, NEG_INF);
  fill_f32<<<cdiv(n, 256), 256, 0, stream>>>(zz, n, 0.0f);
  fill_f32<<<cdiv((long)n * HC, 256), 256, 0, stream>>>(out, n * HC, 0.0f);
  node_alpha<<<cdiv((long)n * 32, 256), 256, 0, stream>>>(h2, as2, ad2, aS, aD, n, HC);
  edge_softmax_max<<<em_blocks, 256, 0, stream>>>(srcp, dstp, aS, aD, mx, ne, n);
  edge_softmax_sum<<<em_blocks, 256, 0, stream>>>(srcp, dstp, aS, aD, mx, zz, ne, n);
  edge_aggregate<<<cdiv((long)(ne + n) << 7, 256), 256, 0, stream>>>(
      srcp, dstp, aS, aD, mx, zz, h2, out, ne, n, 7);

  // ---- Fused head: out = x@Wl1^T + bl1 + relu((agg2+b2)@Wl2^T + bl2) ----
  final_fused_wmma<<<cdiv(n, 16), 256, 0, stream>>>(x, b2, Wl1, bl1, Wl2, bl2, out, n);
}